// Classifier_39067022525085
// MI455X (gfx1250) — compile-verified
//
#include <hip/hip_runtime.h>
#include <math.h>

typedef __attribute__((ext_vector_type(2))) float v2f;
typedef __attribute__((ext_vector_type(8))) float v8f;

// ---------------------------------------------------------------------------
// Zero-fill (grid-stride)
// ---------------------------------------------------------------------------
__global__ void fill_zero_kernel(float* __restrict__ p, size_t n) {
    size_t i = (size_t)blockIdx.x * blockDim.x + threadIdx.x;
    size_t stride = (size_t)gridDim.x * blockDim.x;
    for (; i < n; i += stride) p[i] = 0.0f;
}

// ---------------------------------------------------------------------------
// In-degree per destination node (computed once, reused by both layers)
// ---------------------------------------------------------------------------
__global__ void edge_degree_kernel(const int* __restrict__ dst,
                                   float* __restrict__ deg, int n_edges) {
    int e = blockIdx.x * blockDim.x + threadIdx.x;
    if (e < n_edges) atomicAdd(&deg[dst[e]], 1.0f);
}

// ---------------------------------------------------------------------------
// Node feature transform Y = X @ W using fp32 WMMA (V_WMMA_F32_16X16X4_F32).
// One wave handles a tile of 16 nodes -> 16x16 fp32 accumulator (8 VGPRs).
// A layout: lane L holds row (L%16), K-pair offset (L<16 ? 0 : 2).
// B layout: lane L holds col (L%16), same K-pair split.
// D layout: VGPR r holds row r + (L<16 ? 0 : 8), col (L%16).
// K is padded up to KPAD (multiple of 4). Tail masking is branchless (clamped
// address + value select), and the store epilogue bound-check is hoisted into
// a wave-uniform branch so full tiles (all of them when N%16==0) issue eight
// unguarded stores with no exec-mask round-trips.
// ---------------------------------------------------------------------------
template <int K, int KPAD>
__global__ void node_gemm_wmma_kernel(const float* __restrict__ X,
                                      const float* __restrict__ W,
                                      float* __restrict__ Y,
                                      int n_nodes) {
    int gwave  = (blockIdx.x * blockDim.x + threadIdx.x) >> 5;
    int lane   = threadIdx.x & 31;
    int ntiles = (n_nodes + 15) >> 4;
    if (gwave >= ntiles) return;                 // wave-uniform: EXEC stays full

    int m    = lane & 15;                        // row within tile (A) / col (B,D)
    int node = gwave * 16 + m;
    if (node >= n_nodes) node = n_nodes - 1;     // clamp partial tile (dupe rows)
    int koff = (lane >> 4) << 1;                 // 0 for lanes 0-15, 2 for 16-31
    int col  = m;

    const float* xrow = X + (size_t)node * K;
    v8f acc = {};
#pragma unroll
    for (int k0 = 0; k0 < KPAD; k0 += 4) {
        int ka = k0 + koff;
        // branchless tail masking: clamp address, zero value
        int  k0i = (ka + 0 < K) ? (ka + 0) : (K - 1);
        int  k1i = (ka + 1 < K) ? (ka + 1) : (K - 1);
        float m0 = (ka + 0 < K) ? 1.0f : 0.0f;
        float m1 = (ka + 1 < K) ? 1.0f : 0.0f;
        v2f a, b;
        a.x = xrow[k0i] * m0;
        a.y = xrow[k1i] * m1;
        b.x = W[k0i * 16 + col] * m0;
        b.y = W[k1i * 16 + col] * m1;
        acc = __builtin_amdgcn_wmma_f32_16x16x4_f32(
            /*neg_a=*/false, a, /*neg_b=*/false, b,
            /*c_mod=*/(short)0, acc, /*reuse_a=*/false, /*reuse_b=*/false);
    }

    int rowbase = gwave * 16 + ((lane >> 4) << 3);
    if (gwave * 16 + 16 <= n_nodes) {
        // wave-uniform fast path: full tile, no bounds checks
        float* yout = Y + (size_t)rowbase * 16 + col;
#pragma unroll
        for (int r = 0; r < 8; ++r) {
            yout[r * 16] = acc[r];
        }
    } else {
        // final partial tile only
#pragma unroll
        for (int r = 0; r < 8; ++r) {
            int row = rowbase + r;
            if (row < n_nodes) Y[(size_t)row * 16 + col] = acc[r];
        }
    }
}

// ---------------------------------------------------------------------------
// Edge scatter: agg[dst] += Y[src] (16 fp32 channels). Y and agg both live in
// L2 (64 MB each vs 192 MB L2), so random access stays on-chip.
// ---------------------------------------------------------------------------
__global__ void edge_scatter_kernel(const float* __restrict__ Y,
                                    const int* __restrict__ src,
                                    const int* __restrict__ dst,
                                    float* __restrict__ agg, int n_edges) {
    int e = blockIdx.x * blockDim.x + threadIdx.x;
    if (e >= n_edges) return;
    int s = src[e];
    int d = dst[e];
    const float4* ys = (const float4*)(Y + (size_t)s * 16);
    float* ad = agg + (size_t)d * 16;
#pragma unroll
    for (int q = 0; q < 4; ++q) {
        float4 v = ys[q];
        atomicAdd(ad + 4 * q + 0, v.x);
        atomicAdd(ad + 4 * q + 1, v.y);
        atomicAdd(ad + 4 * q + 2, v.z);
        atomicAdd(ad + 4 * q + 3, v.w);
    }
}

// ---------------------------------------------------------------------------
// Combine: x = relu( (deg>0 ? agg/deg : y) + bias ), in-place into agg buffer.
// (Valid because segment-mean commutes with the linear projection.)
// ---------------------------------------------------------------------------
__global__ void combine_relu_kernel(const float* __restrict__ Y,
                                    const float* __restrict__ deg,
                                    const float* __restrict__ bias,
                                    float* __restrict__ A, int n_nodes) {
    int i = blockIdx.x * blockDim.x + threadIdx.x;
    if (i >= n_nodes) return;
    float dm   = deg[i];
    bool  mail = dm > 0.0f;
    float inv  = mail ? (1.0f / dm) : 0.0f;
    const float* y = Y + (size_t)i * 16;
    float*       a = A + (size_t)i * 16;
#pragma unroll
    for (int c = 0; c < 16; ++c) {
        float base = mail ? a[c] * inv : y[c];
        float v = base + bias[c];
        a[c] = v > 0.0f ? v : 0.0f;
    }
}

// ---------------------------------------------------------------------------
// Per-graph sum pooling + counts (mean taken in the head).
// ---------------------------------------------------------------------------
__global__ void pool_kernel(const float* __restrict__ X,
                            const int* __restrict__ gid,
                            float* __restrict__ gsum,
                            float* __restrict__ gcnt, int n_nodes) {
    int i = blockIdx.x * blockDim.x + threadIdx.x;
    if (i >= n_nodes) return;
    int g = gid[i];
    const float* x  = X + (size_t)i * 16;
    float*       gs = gsum + (size_t)g * 16;
#pragma unroll
    for (int c = 0; c < 16; ++c) atomicAdd(gs + c, x[c]);
    atomicAdd(gcnt + g, 1.0f);
}

// ---------------------------------------------------------------------------
// MLP head: hg = gsum/max(cnt,1); z = (hg@Wf1+bf1)@Wf2+bf2; out = sigmoid(z)
// ---------------------------------------------------------------------------
__global__ void head_kernel(const float* __restrict__ gsum,
                            const float* __restrict__ gcnt,
                            const float* __restrict__ Wf1,
                            const float* __restrict__ bf1,
                            const float* __restrict__ Wf2,
                            const float* __restrict__ bf2,
                            float* __restrict__ out, int G) {
    int g = blockIdx.x * blockDim.x + threadIdx.x;
    if (g >= G) return;
    float inv = 1.0f / fmaxf(gcnt[g], 1.0f);
    float hg[16];
#pragma unroll
    for (int k = 0; k < 16; ++k) hg[k] = gsum[g * 16 + k] * inv;
    float z = bf2[0];
#pragma unroll
    for (int j = 0; j < 8; ++j) {
        float t = bf1[j];
#pragma unroll
        for (int k = 0; k < 16; ++k) t += hg[k] * Wf1[k * 8 + j];
        z += t * Wf2[j];
    }
    out[g] = 1.0f / (1.0f + expf(-z));
}

// ---------------------------------------------------------------------------
// Launch
// ---------------------------------------------------------------------------
extern "C" void kernel_launch(void* const* d_in, const int* in_sizes, int n_in,
                              void* d_out, int out_size, void* d_ws, size_t ws_size,
                              hipStream_t stream) {
    const float* h    = (const float*)d_in[0];
    const int*   src  = (const int*)d_in[1];
    const int*   dst  = (const int*)d_in[2];
    const int*   gid  = (const int*)d_in[3];
    const float* W1   = (const float*)d_in[4];
    const float* b1   = (const float*)d_in[5];
    const float* W2   = (const float*)d_in[6];
    const float* b2   = (const float*)d_in[7];
    const float* Wf1  = (const float*)d_in[8];
    const float* bf1  = (const float*)d_in[9];
    const float* Wf2  = (const float*)d_in[10];
    const float* bf2  = (const float*)d_in[11];
    float* out = (float*)d_out;

    const int IN_FEATS = 22;
    int N = in_sizes[0] / IN_FEATS;   // 1,000,000
    int E = in_sizes[1];              // 16,000,000
    int G = out_size;                 // 256

    // Workspace layout: deg[N] | bufA[N*16] | bufB[N*16] | gsum[G*16] | gcnt[G]
    float* deg  = (float*)d_ws;
    float* bufA = deg + N;
    float* bufB = bufA + (size_t)N * 16;
    float* gsum = bufB + (size_t)N * 16;
    float* gcnt = gsum + (size_t)G * 16;

    int edgeBlocks = (E + 255) / 256;
    int nodeBlocks = (N + 255) / 256;
    int tiles      = (N + 15) / 16;
    int gemmBlocks = (tiles + 7) / 8;   // 8 waves (256 threads) per block

    // init: deg, agg, pooling accumulators
    fill_zero_kernel<<<1024, 256, 0, stream>>>(deg, (size_t)N);
    fill_zero_kernel<<<4096, 256, 0, stream>>>(bufB, (size_t)N * 16);
    fill_zero_kernel<<<32, 256, 0, stream>>>(gsum, (size_t)G * 17); // gsum+gcnt

    edge_degree_kernel<<<edgeBlocks, 256, 0, stream>>>(dst, deg, E);

    // ---- layer 1: project (WMMA) -> scatter -> combine ----
    node_gemm_wmma_kernel<22, 24><<<gemmBlocks, 256, 0, stream>>>(h, W1, bufA, N);
    edge_scatter_kernel<<<edgeBlocks, 256, 0, stream>>>(bufA, src, dst, bufB, E);
    combine_relu_kernel<<<nodeBlocks, 256, 0, stream>>>(bufA, deg, b1, bufB, N);

    // ---- layer 2: project (WMMA) -> scatter -> combine ----
    node_gemm_wmma_kernel<16, 16><<<gemmBlocks, 256, 0, stream>>>(bufB, W2, bufA, N);
    fill_zero_kernel<<<4096, 256, 0, stream>>>(bufB, (size_t)N * 16);
    edge_scatter_kernel<<<edgeBlocks, 256, 0, stream>>>(bufA, src, dst, bufB, E);
    combine_relu_kernel<<<nodeBlocks, 256, 0, stream>>>(bufA, deg, b2, bufB, N);

    // ---- pooling + head ----
    pool_kernel<<<nodeBlocks, 256, 0, stream>>>(bufB, gid, gsum, gcnt, N);
    head_kernel<<<(G + 63) / 64, 64, 0, stream>>>(gsum, gcnt, Wf1, bf1, Wf2, bf2, out, G);
}